// EEGPreprocessor_26749056319877
// MI455X (gfx1250) — compile-verified
//
#include <hip/hip_runtime.h>
#include <math.h>

// EEG preprocessor for MI455X (gfx1250, wave32).
// x: (256, 128, 2048) f32. Memory-bound: 512 MB min HBM traffic -> ~22us @ 23.3 TB/s.
// One block per batch (1 MB slab) so passes 2/3 re-read from L2, not HBM.
// Channel-sum pass is expressed as ones(16x4) x X(4x16) chains on
// V_WMMA_F32_16X16X4_F32 (f32 WMMA keeps normalization numerics sane).

typedef float v2f __attribute__((ext_vector_type(2)));
typedef float v8f __attribute__((ext_vector_type(8)));

#define NB 256
#define NC 128
#define NT 2048
#define SLAB (NC * NT)
#define NTHREADS 512
#define NWAVES (NTHREADS / 32)   // 16 waves

__global__ __launch_bounds__(NTHREADS)
void eeg_preprocess_kernel(const float* __restrict__ x, float* __restrict__ out) {
    __shared__ float s_m[NT];      // per-time channel mean (8 KB)
    __shared__ float s_mean[NC];   // per-channel mean of centered data
    __shared__ float s_istd[NC];   // per-channel 1/std (std==0 -> 1)

    const int b    = blockIdx.x;
    const float* __restrict__ xb = x   + (size_t)b * SLAB;
    float* __restrict__       ob = out + (size_t)b * SLAB;
    const int tid  = threadIdx.x;
    const int wave = tid >> 5;
    const int lane = tid & 31;

    // ---------------- Pass 1: S[t] = sum_c x[c,t] via WMMA f32 16x16x4 ----
    // A = ones(16x4). D[m,n] = sum_k B[k,n]  (every row of D is the column sum,
    // so any K-row permutation of B is harmless). Chain 32 WMMAs to cover K=128.
    {
        const int col  = lane & 15;      // N index (matches C/D lane layout)
        const int half = lane >> 4;      // B half: lanes 0-15 vs 16-31
        const int colsPerWave = NT / NWAVES;          // 128 time columns
        const v2f a = {1.0f, 1.0f};                   // all-ones A block

        for (int j = 0; j < colsPerWave / 16; ++j) {  // 8 tiles of 16 columns
            const int t0 = wave * colsPerWave + j * 16;
            v8f acc = {0.f, 0.f, 0.f, 0.f, 0.f, 0.f, 0.f, 0.f};
            #pragma unroll 4
            for (int ks = 0; ks < NC / 4; ++ks) {     // K = 4 channels per step
                const int cbase = ks * 4;
                const int r0 = cbase + (half ? 2 : 0);  // VGPR0: K0(lo)/K2(hi)
                const int r1 = r0 + 1;                  // VGPR1: K1(lo)/K3(hi)
                v2f bm;
                bm.x = xb[r0 * NT + t0 + col];
                bm.y = xb[r1 * NT + t0 + col];
                acc = __builtin_amdgcn_wmma_f32_16x16x4_f32(
                    /*neg_a=*/false, a, /*neg_b=*/false, bm,
                    /*c_mod=*/(short)0, acc, /*reuse_a=*/false, /*reuse_b=*/false);
            }
            // D row M=0 lives in acc[0], lanes 0-15, N = lane.
            if (lane < 16) s_m[t0 + lane] = acc[0] * (1.0f / (float)NC);
        }
    }
    __syncthreads();

    // ---------------- Pass 2: per-channel mean/istd of y = x - m ----------
    {
        const int chPerWave = NC / NWAVES;            // 8 channels per wave
        for (int jc = 0; jc < chPerWave; ++jc) {
            const int c = wave * chPerWave + jc;
            const float* __restrict__ row = xb + c * NT;
            float s1 = 0.f, s2 = 0.f;
            for (int t = lane; t < NT; t += 32) {     // coalesced 32-lane stream
                const float d = row[t] - s_m[t];
                s1 += d;
                s2 += d * d;
            }
            #pragma unroll
            for (int off = 16; off > 0; off >>= 1) {  // wave32 tree reduction
                s1 += __shfl_xor(s1, off, 32);
                s2 += __shfl_xor(s2, off, 32);
            }
            if (lane == 0) {
                const float mean = s1 * (1.0f / (float)NT);
                const float var  = fmaxf(s2 * (1.0f / (float)NT) - mean * mean, 0.0f);
                const float sd   = sqrtf(var);
                s_mean[c] = mean;
                s_istd[c] = (sd > 0.0f) ? (1.0f / sd) : 1.0f;  // std==0 -> 1
            }
        }
    }
    __syncthreads();

    // ---------------- Pass 3: streamed normalize (L2-hot reread) ----------
    for (int i = tid; i < SLAB; i += NTHREADS) {
        const int c = i >> 11;          // / NT
        const int t = i & (NT - 1);
        __builtin_prefetch(xb + i + 8 * NTHREADS, 0, 0);  // global_prefetch hint
        ob[i] = (xb[i] - s_m[t] - s_mean[c]) * s_istd[c];
    }
}

extern "C" void kernel_launch(void* const* d_in, const int* in_sizes, int n_in,
                              void* d_out, int out_size, void* d_ws, size_t ws_size,
                              hipStream_t stream) {
    const float* x = (const float*)d_in[0];
    float* out     = (float*)d_out;
    (void)in_sizes; (void)n_in; (void)out_size; (void)d_ws; (void)ws_size;
    eeg_preprocess_kernel<<<NB, NTHREADS, 0, stream>>>(x, out);
}